// DeepGCNConv_8744553414739
// MI455X (gfx1250) — compile-verified
//
#include <hip/hip_runtime.h>
#include <hip/hip_bf16.h>

typedef float v2f __attribute__((ext_vector_type(2)));
typedef float v8f __attribute__((ext_vector_type(8)));

#define DFEAT   128
#define NGRAPH  64
#define NCLS    16

// ---------------------------------------------------------------------------
// deg[i] = 1 (self loop); then atomic in-degree over edges; dinv = rsqrt(deg)
// ---------------------------------------------------------------------------
__global__ void init_deg_kernel(float* deg, int n) {
    int i = blockIdx.x * blockDim.x + threadIdx.x;
    if (i < n) deg[i] = 1.0f;   // self-loop contribution
}

__global__ void accum_deg_kernel(const int* __restrict__ dst, float* deg, int E) {
    int e = blockIdx.x * blockDim.x + threadIdx.x;
    if (e < E) atomicAdd(&deg[dst[e]], 1.0f);
}

__global__ void dinv_kernel(const float* __restrict__ deg, float* dinv, int n) {
    int i = blockIdx.x * blockDim.x + threadIdx.x;
    if (i < n) dinv[i] = __frsqrt_rn(deg[i]);   // deg >= 1 always
}

// ---------------------------------------------------------------------------
// H = X (MxK) @ W (KxN), fp32 WMMA 16x16x4.
// One wave computes a full 16x128 strip of H: A is loaded once per K-step and
// reused across 8 N-tiles (8 independent v_wmma per step -> good co-execution,
// 8x fewer X reads than a 16x16-per-wave scheme).
//   A 16x4 f32 layout: lane l holds row M=l%16, K = 2*(l/16)+{0,1}
//   B 4x16 f32 layout: lane l holds col N=l%16, K = 2*(l/16)+{0,1}
//   C/D 16x16 f32:     VGPR i holds M = i + 8*(l/16), N = l%16
// Requires M % 16 == 0, K == N == 128.
// ---------------------------------------------------------------------------
__global__ __launch_bounds__(32)
void gemm_f32_wmma_kernel(const float* __restrict__ X,
                          const float* __restrict__ W,
                          float* __restrict__ H) {
    const int lane = threadIdx.x;       // 0..31 (wave32)
    const int half = lane >> 4;         // 0 | 1
    const int l16  = lane & 15;
    const int m0   = blockIdx.x * 16;
    const int K = DFEAT, N = DFEAT;

    const float* xrow = X + (size_t)(m0 + l16) * K;

    v8f acc[8];
#pragma unroll
    for (int j = 0; j < 8; ++j) acc[j] = (v8f){};

    for (int k0 = 0; k0 < K; k0 += 4) {
        const int ka = k0 + half * 2;
        v2f a;
        a.x = xrow[ka];
        a.y = xrow[ka + 1];
        const float* w0 = W + (size_t)ka * N + l16;        // row ka
        const float* w1 = w0 + N;                          // row ka+1
#pragma unroll
        for (int j = 0; j < 8; ++j) {
            v2f b;
            b.x = w0[j * 16];
            b.y = w1[j * 16];
            // 8-arg form: (neg_a, A, neg_b, B, c_mod, C, reuse_a, reuse_b)
            acc[j] = __builtin_amdgcn_wmma_f32_16x16x4_f32(
                false, a, false, b, (short)0, acc[j], false, false);
        }
    }

#pragma unroll
    for (int j = 0; j < 8; ++j) {
#pragma unroll
        for (int i = 0; i < 8; ++i) {
            const int row = m0 + i + half * 8;
            H[(size_t)row * N + j * 16 + l16] = acc[j][i];
        }
    }
}

// ---------------------------------------------------------------------------
// Seed accumulator with self-loop + bias: Out[i,f] = H[i,f]*dinv[i]^2 + b[f]
// (plain stores; runs before the edge scatter)
// ---------------------------------------------------------------------------
__global__ void seed_out_kernel(const float* __restrict__ H,
                                const float* __restrict__ dinv,
                                const float* __restrict__ bias,
                                float* __restrict__ Out, size_t total) {
    size_t i = (size_t)blockIdx.x * blockDim.x + threadIdx.x;
    if (i >= total) return;
    const int node = (int)(i >> 7);     // /128
    const int f    = (int)(i & 127);
    const float di = dinv[node];
    Out[i] = H[i] * di * di + bias[f];
}

// ---------------------------------------------------------------------------
// Edge scatter: Out[dst] += H[src] * (dinv[src]*dinv[dst])
// One wave per edge; lane handles 4 consecutive features (32*4 = 128).
// ---------------------------------------------------------------------------
__global__ __launch_bounds__(256)
void scatter_edges_kernel(const float* __restrict__ H,
                          const int* __restrict__ src,
                          const int* __restrict__ dst,
                          const float* __restrict__ dinv,
                          float* __restrict__ Out, int E) {
    const int wavesPerBlock = blockDim.x >> 5;
    const int wid  = threadIdx.x >> 5;
    const int lane = threadIdx.x & 31;
    const int e = blockIdx.x * wavesPerBlock + wid;
    if (e >= E) return;

    const int s = src[e];
    const int d = dst[e];

    // Prefetch next edge's source row into cache (global_prefetch_b8 path).
    if (e + wavesPerBlock < E) {
        const int sn = src[e + wavesPerBlock];
        __builtin_prefetch(H + (size_t)sn * DFEAT + lane * 4, 0, 3);
    }

    const float norm = dinv[s] * dinv[d];
    const float4 v = ((const float4*)(H + (size_t)s * DFEAT))[lane];
    float* orow = Out + (size_t)d * DFEAT + lane * 4;
    atomicAdd(orow + 0, v.x * norm);
    atomicAdd(orow + 1, v.y * norm);
    atomicAdd(orow + 2, v.z * norm);
    atomicAdd(orow + 3, v.w * norm);
}

// ---------------------------------------------------------------------------
// In-place ReLU
// ---------------------------------------------------------------------------
__global__ void relu_kernel(float* __restrict__ p, size_t n) {
    size_t i = (size_t)blockIdx.x * blockDim.x + threadIdx.x;
    if (i < n) p[i] = fmaxf(p[i], 0.0f);
}

// ---------------------------------------------------------------------------
// Zero a float buffer
// ---------------------------------------------------------------------------
__global__ void zero_kernel(float* p, size_t n) {
    size_t i = (size_t)blockIdx.x * blockDim.x + threadIdx.x;
    if (i < n) p[i] = 0.0f;
}

// ---------------------------------------------------------------------------
// Mean pool over batch: pool[g,f] += h[node,f]; cnt[g] += 1 (once per node)
// ---------------------------------------------------------------------------
__global__ void pool_kernel(const float* __restrict__ h,
                            const int* __restrict__ batch,
                            float* __restrict__ pool,
                            float* __restrict__ cnt, size_t total) {
    size_t i = (size_t)blockIdx.x * blockDim.x + threadIdx.x;
    if (i >= total) return;
    const int node = (int)(i >> 7);
    const int f    = (int)(i & 127);
    const int g    = batch[node];
    atomicAdd(&pool[(size_t)g * DFEAT + f], h[i]);
    if (f == 0) atomicAdd(&cnt[g], 1.0f);
}

// ---------------------------------------------------------------------------
// Final classifier: out[g,c] = (pool[g,:] @ lin_w[:,c]) / max(cnt[g],1) + lb[c]
// ---------------------------------------------------------------------------
__global__ void final_linear_kernel(const float* __restrict__ pool,
                                    const float* __restrict__ cnt,
                                    const float* __restrict__ lw,
                                    const float* __restrict__ lb,
                                    float* __restrict__ out) {
    int i = blockIdx.x * blockDim.x + threadIdx.x;
    if (i >= NGRAPH * NCLS) return;
    const int g = i / NCLS;
    const int c = i % NCLS;
    float s = 0.0f;
    for (int f = 0; f < DFEAT; ++f)
        s += pool[(size_t)g * DFEAT + f] * lw[(size_t)f * NCLS + c];
    out[i] = s / fmaxf(cnt[g], 1.0f) + lb[c];
}

// ---------------------------------------------------------------------------
extern "C" void kernel_launch(void* const* d_in, const int* in_sizes, int n_in,
                              void* d_out, int out_size, void* d_ws, size_t ws_size,
                              hipStream_t stream) {
    const float* x     = (const float*)d_in[0];
    const int*   edges = (const int*)d_in[1];      // [2, E] flat
    const int*   batch = (const int*)d_in[2];
    const float* W1 = (const float*)d_in[3];  const float* b1 = (const float*)d_in[4];
    const float* W2 = (const float*)d_in[5];  const float* b2 = (const float*)d_in[6];
    const float* W3 = (const float*)d_in[7];  const float* b3 = (const float*)d_in[8];
    const float* lw = (const float*)d_in[9];  const float* lb = (const float*)d_in[10];
    float* out = (float*)d_out;

    const int N = in_sizes[0] / DFEAT;             // 50000
    const int E = in_sizes[1] / 2;                 // 800000
    const int* src = edges;
    const int* dst = edges + E;
    const size_t NF = (size_t)N * DFEAT;

    // Workspace partition
    float* deg  = (float*)d_ws;                    // N
    float* dinv = deg  + N;                        // N
    float* Hbuf = dinv + N;                        // N*128  (GEMM temp)
    float* Abuf = Hbuf + NF;                       // N*128  (layer out ping)
    float* Bbuf = Abuf + NF;                       // N*128  (layer out pong)
    float* pool = Bbuf + NF;                       // 64*128
    // cnt = pool + NGRAPH*DFEAT                   // 64 (zeroed together)

    const int T = 256;
    // --- degrees / norms ---
    init_deg_kernel<<<(N + T - 1) / T, T, 0, stream>>>(deg, N);
    accum_deg_kernel<<<(E + T - 1) / T, T, 0, stream>>>(dst, deg, E);
    dinv_kernel<<<(N + T - 1) / T, T, 0, stream>>>(deg, dinv, N);

    const int gemmGrid = N / 16;                   // 3125 strips, exact
    const int scatterBlocks = (E + 7) / 8;         // 8 waves/block, 1 edge/wave
    const int nfBlocks = (int)((NF + T - 1) / T);

    // --- layer 1: x -> Abuf (ReLU) ---
    gemm_f32_wmma_kernel<<<gemmGrid, 32, 0, stream>>>(x, W1, Hbuf);
    seed_out_kernel<<<nfBlocks, T, 0, stream>>>(Hbuf, dinv, b1, Abuf, NF);
    scatter_edges_kernel<<<scatterBlocks, 256, 0, stream>>>(Hbuf, src, dst, dinv, Abuf, E);
    relu_kernel<<<nfBlocks, T, 0, stream>>>(Abuf, NF);

    // --- layer 2: Abuf -> Bbuf (ReLU) ---
    gemm_f32_wmma_kernel<<<gemmGrid, 32, 0, stream>>>(Abuf, W2, Hbuf);
    seed_out_kernel<<<nfBlocks, T, 0, stream>>>(Hbuf, dinv, b2, Bbuf, NF);
    scatter_edges_kernel<<<scatterBlocks, 256, 0, stream>>>(Hbuf, src, dst, dinv, Bbuf, E);
    relu_kernel<<<nfBlocks, T, 0, stream>>>(Bbuf, NF);

    // --- layer 3: Bbuf -> Abuf (no ReLU) ---
    gemm_f32_wmma_kernel<<<gemmGrid, 32, 0, stream>>>(Bbuf, W3, Hbuf);
    seed_out_kernel<<<nfBlocks, T, 0, stream>>>(Hbuf, dinv, b3, Abuf, NF);
    scatter_edges_kernel<<<scatterBlocks, 256, 0, stream>>>(Hbuf, src, dst, dinv, Abuf, E);

    // --- mean pool + classifier ---
    const size_t poolN = (size_t)NGRAPH * DFEAT + NGRAPH;   // pool + cnt
    zero_kernel<<<(int)((poolN + T - 1) / T), T, 0, stream>>>(pool, poolN);
    pool_kernel<<<nfBlocks, T, 0, stream>>>(Abuf, batch, pool, pool + (size_t)NGRAPH * DFEAT, NF);
    final_linear_kernel<<<(NGRAPH * NCLS + T - 1) / T, T, 0, stream>>>(
        pool, pool + (size_t)NGRAPH * DFEAT, lw, lb, out);
}